// MMD_loss_5368709120075
// MI455X (gfx1250) — compile-verified
//
#include <hip/hip_runtime.h>

typedef __attribute__((ext_vector_type(2))) float v2f;
typedef __attribute__((ext_vector_type(8))) float v8f;

#define NROWS 8192
#define HALF  4096
#define KDIM  256
#define MT    128      // macro tile (rows/cols per block)
#define KC    16       // K chunk staged in LDS (double-buffered)
#define NCHUNK (KDIM / KC)
#define LDSS  20       // padded LDS row stride in floats (bank-conflict-free)
#define BUFF  (MT * LDSS)   // floats per buffer per matrix
#define NBLK  64       // 8192 / 128
#define NPART (NBLK * NBLK)

__device__ __forceinline__ const float* row_base(const float* __restrict__ s,
                                                 const float* __restrict__ t,
                                                 int row) {
  return (row < HALF) ? (s + (size_t)row * KDIM)
                      : (t + (size_t)(row - HALF) * KDIM);
}

// CDNA5 async DMA: global -> LDS, 16 bytes per lane, tracked by ASYNCcnt.
__device__ __forceinline__ void async_b128(const float* g, unsigned lds_addr) {
  asm volatile("global_load_async_to_lds_b128 %0, %1, off"
               :: "v"(lds_addr), "v"(g)
               : "memory");
}
__device__ __forceinline__ void wait_async_4() {
  asm volatile("s_wait_asynccnt 0x4" ::: "memory");
}
__device__ __forceinline__ void wait_async_0() {
  asm volatile("s_wait_asynccnt 0x0" ::: "memory");
}

// ---- per-row squared norms: one block per row ----
__global__ void sq_kernel(const float* __restrict__ src,
                          const float* __restrict__ tgt,
                          float* __restrict__ sq) {
  __shared__ float red[256];
  const int row = blockIdx.x;
  const float* p = row_base(src, tgt, row);
  float v = p[threadIdx.x];
  red[threadIdx.x] = v * v;
  __syncthreads();
  for (int s = 128; s > 0; s >>= 1) {
    if (threadIdx.x < (unsigned)s) red[threadIdx.x] += red[threadIdx.x + s];
    __syncthreads();
  }
  if (threadIdx.x == 0) sq[row] = red[0];
}

// ---- main tiled kernel: MODE 0 -> sum(max(d2,0)); MODE 1 -> signed kernel sum ----
// Only lower-triangle blocks (bj <= bi) compute; off-diagonal blocks weighted x2.
// Waves tiled 4x2: each wave computes a 32x64 sub-tile (2 M-tiles x 4 N-tiles).
template <int MODE>
__global__ __launch_bounds__(256) void mmd_tile_kernel(
    const float* __restrict__ src, const float* __restrict__ tgt,
    const float* __restrict__ sq, float* __restrict__ part,
    const float* __restrict__ coef) {
  __shared__ float As[2 * BUFF];   // double-buffered A chunk
  __shared__ float Bs[2 * BUFF];   // double-buffered B chunk
  __shared__ float red[256];

  const int tid = threadIdx.x;
  const int bi  = blockIdx.y;
  const int bj  = blockIdx.x;

  if (bj > bi) {                       // symmetric half skipped
    if (tid == 0) part[bi * NBLK + bj] = 0.0f;
    return;
  }

  const int ibase = bi * MT;
  const int jbase = bj * MT;
  const int wave   = tid >> 5;
  const int lane   = tid & 31;
  const int m      = lane & 15;
  const int hi     = lane >> 4;        // 0 or 1
  const int colPar = hi * 2;           // K offset within the 4-wide slice
  const int mstrip = wave >> 1;        // 0..3: which 32-row strip
  const int nhalf  = wave & 1;         // 0..1: which 64-col half

  // Staging geometry: chunk = 128 rows x 16 floats = 512 float4; 2 per thread.
  // A 128-row macro tile never straddles the source/target boundary.
  const int r0 = tid >> 2;             // 0..63
  const int c0 = (tid & 3) << 2;       // 0,4,8,12
  const float* gA = row_base(src, tgt, ibase) + (size_t)r0 * KDIM + c0;
  const float* gB = row_base(src, tgt, jbase) + (size_t)r0 * KDIM + c0;
  const unsigned ldsA0 = (unsigned)(uintptr_t)(&As[0]) + (unsigned)(r0 * LDSS + c0) * 4u;
  const unsigned ldsB0 = (unsigned)(uintptr_t)(&Bs[0]) + (unsigned)(r0 * LDSS + c0) * 4u;

  // issue chunk 0 DMA into buffer 0
#pragma unroll
  for (int t = 0; t < 2; ++t) {
    async_b128(gA + t * 64 * KDIM, ldsA0 + (unsigned)(t * 64 * LDSS) * 4u);
    async_b128(gB + t * 64 * KDIM, ldsB0 + (unsigned)(t * 64 * LDSS) * 4u);
  }

  v8f acc[2][4] = {};                  // [it][jt]: 2x4 grid of 16x16 f32 tiles
  const int aoff = (mstrip * 32 + m) * LDSS + colPar;       // + it*16*LDSS
  const int boff = (nhalf * 64 + m) * LDSS + colPar;        // + jt*16*LDSS

#pragma unroll
  for (int chunk = 0; chunk < NCHUNK; ++chunk) {
    const int cur = chunk & 1;
    if (chunk + 1 < NCHUNK) {
      // buffer (1-cur) was released by the barrier ending chunk-1's compute
      const int kc  = (chunk + 1) * KC;
      const unsigned bofs = (unsigned)((1 - cur) * BUFF) * 4u;
#pragma unroll
      for (int t = 0; t < 2; ++t) {
        async_b128(gA + t * 64 * KDIM + kc, ldsA0 + bofs + (unsigned)(t * 64 * LDSS) * 4u);
        async_b128(gB + t * 64 * KDIM + kc, ldsB0 + bofs + (unsigned)(t * 64 * LDSS) * 4u);
      }
      wait_async_4();                  // chunk's 4 loads done; next 4 may fly
    } else {
      wait_async_0();
    }
    __syncthreads();                   // all waves' chunk data visible in LDS

    const float* Acur = &As[cur * BUFF];
    const float* Bcur = &Bs[cur * BUFF];
#pragma unroll
    for (int kk = 0; kk < KC; kk += 8) {
      v2f a0[2], a1[2], b0[4], b1[4];
#pragma unroll
      for (int it = 0; it < 2; ++it) {
        a0[it] = *(const v2f*)(&Acur[aoff + it * 16 * LDSS + kk]);
        a1[it] = *(const v2f*)(&Acur[aoff + it * 16 * LDSS + kk + 4]);
      }
#pragma unroll
      for (int jt = 0; jt < 4; ++jt) {
        b0[jt] = *(const v2f*)(&Bcur[boff + jt * 16 * LDSS + kk]);
        b1[jt] = *(const v2f*)(&Bcur[boff + jt * 16 * LDSS + kk + 4]);
      }
#pragma unroll
      for (int it = 0; it < 2; ++it)
#pragma unroll
        for (int jt = 0; jt < 4; ++jt) {
          acc[it][jt] = __builtin_amdgcn_wmma_f32_16x16x4_f32(
              false, a0[it], false, b0[jt], (short)0, acc[it][jt], false, false);
          acc[it][jt] = __builtin_amdgcn_wmma_f32_16x16x4_f32(
              false, a1[it], false, b1[jt], (short)0, acc[it][jt], false, false);
        }
    }
    __syncthreads();                   // release buffer `cur` for DMA reuse
  }

  // ---- epilogue: d2 = sq_i + sq_j - 2*G, clamp, accumulate ----
  float c = 0.0f, sgn = 1.0f;
  if (MODE == 1) {
    c   = coef[0];  // -1 / (16 * bandwidth)
    sgn = ((ibase < HALF) == (jbase < HALF)) ? 1.0f : -1.0f;
  }

  float local = 0.0f;
#pragma unroll
  for (int it = 0; it < 2; ++it) {
    float sqi[8];
#pragma unroll
    for (int e = 0; e < 8; ++e)
      sqi[e] = sq[ibase + mstrip * 32 + it * 16 + e + hi * 8];
#pragma unroll
    for (int jt = 0; jt < 4; ++jt) {
      float sqj = sq[jbase + nhalf * 64 + jt * 16 + m];
#pragma unroll
      for (int e = 0; e < 8; ++e) {
        float d2 = sqi[e] + sqj - 2.0f * acc[it][jt][e];
        d2 = fmaxf(d2, 0.0f);
        if (MODE == 0) {
          local += d2;
        } else {
          float u = __expf(d2 * c);   // exp(-d2/(16*bw))
          float s = u;
          float p = u * u; s += p;    // u^2
          p *= p;          s += p;    // u^4
          p *= p;          s += p;    // u^8
          p *= p;          s += p;    // u^16 = exp(-d2/bw)
          local += sgn * s;
        }
      }
    }
  }

  red[tid] = local;
  __syncthreads();
  for (int s = 128; s > 0; s >>= 1) {
    if (tid < (unsigned)s) red[tid] += red[tid + s];
    __syncthreads();
  }
  if (tid == 0) {
    float w = (bi == bj) ? 1.0f : 2.0f;   // symmetry weight
    part[bi * NBLK + bj] = red[0] * w;
  }
}

// ---- reductions (single block, fixed order -> deterministic) ----
__global__ void reduce_bw_kernel(const float* __restrict__ part,
                                 float* __restrict__ coef) {
  __shared__ float red[256];
  float s = 0.0f;
  for (int i = threadIdx.x; i < NPART; i += 256) s += part[i];
  red[threadIdx.x] = s;
  __syncthreads();
  for (int k = 128; k > 0; k >>= 1) {
    if (threadIdx.x < (unsigned)k) red[threadIdx.x] += red[threadIdx.x + k];
    __syncthreads();
  }
  if (threadIdx.x == 0) {
    float total = red[0];
    float bandwidth = total / 67100672.0f;  // n*n - n, n = 8192
    bandwidth = bandwidth / 4.0f;           // / KERNEL_MUL^(KERNEL_NUM//2)
    coef[0] = -1.0f / (16.0f * bandwidth);
  }
}

__global__ void reduce_out_kernel(const float* __restrict__ part,
                                  float* __restrict__ out) {
  __shared__ float red[256];
  float s = 0.0f;
  for (int i = threadIdx.x; i < NPART; i += 256) s += part[i];
  red[threadIdx.x] = s;
  __syncthreads();
  for (int k = 128; k > 0; k >>= 1) {
    if (threadIdx.x < (unsigned)k) red[threadIdx.x] += red[threadIdx.x + k];
    __syncthreads();
  }
  if (threadIdx.x == 0) out[0] = red[0] / 16777216.0f;  // / b^2, b = 4096
}

extern "C" void kernel_launch(void* const* d_in, const int* in_sizes, int n_in,
                              void* d_out, int out_size, void* d_ws, size_t ws_size,
                              hipStream_t stream) {
  const float* src = (const float*)d_in[0];
  const float* tgt = (const float*)d_in[1];
  float* out = (float*)d_out;

  float* w     = (float*)d_ws;
  float* sq    = w;                        // 8192 floats
  float* part1 = w + 8192;                 // 4096 floats
  float* coef  = w + 8192 + 4096;          // 1 float (padded to 64)
  float* part2 = w + 8192 + 4096 + 64;     // 4096 floats

  sq_kernel<<<NROWS, 256, 0, stream>>>(src, tgt, sq);

  dim3 grid(NBLK, NBLK);
  mmd_tile_kernel<0><<<grid, 256, 0, stream>>>(src, tgt, sq, part1, coef);
  reduce_bw_kernel<<<1, 256, 0, stream>>>(part1, coef);
  mmd_tile_kernel<1><<<grid, 256, 0, stream>>>(src, tgt, sq, part2, coef);
  reduce_out_kernel<<<1, 256, 0, stream>>>(part2, out);
}